// MeanNSE_47553877901938
// MI455X (gfx1250) — compile-verified
//
#include <hip/hip_runtime.h>
#include <stdint.h>

#define NB 671
#define NSTAT 4
#define STATS (NB * NSTAT)            // 2684 floats
#define THREADS 256
#define WAVES 8                       // 256 / wave32
#define WAVE_TILE 128                 // 32 lanes * 4 elems
#define ITERS 16
#define PIPE 3                        // triple-buffer, distance-2 prefetch
#define WAVE_ELEMS (WAVE_TILE * ITERS)      // 2048
#define BLOCK_ELEMS (WAVE_ELEMS * WAVES)    // 16384
#define STAGE_BYTES 2048              // [0,512) pred | [512,1024) true | [1024,2048) basin(i64)
#define EPS 1e-10f

// Low 32 bits of a flat pointer into LDS == byte offset within the wave's LDS
// allocation (ISA 10.2).
__device__ __forceinline__ uint32_t lds_off(const void* p) {
    return (uint32_t)(uintptr_t)p;
}

// CDNA5 async copy: 16B per lane, global -> LDS, tracked with ASYNCcnt.
__device__ __forceinline__ void async_b128(uint32_t loff, const void* gaddr) {
    asm volatile("global_load_async_to_lds_b128 %0, %1, off"
                 :: "v"(loff), "v"((uint64_t)(uintptr_t)gaddr)
                 : "memory");
}

// SoA stat layout: idx = stat*NB + b  -> ds_add_f32 spreads over all 64 banks.
__global__ __launch_bounds__(THREADS)
void MeanNSE_accum_kernel(const float* __restrict__ y_pred,
                          const float* __restrict__ y_true,
                          const long long* __restrict__ basin,
                          float* __restrict__ gstats) {
    __shared__ float s_stats[STATS];
    __shared__ __align__(16) char s_stage[WAVES][PIPE][STAGE_BYTES];

    const int tid  = threadIdx.x;
    const int wave = tid >> 5;
    const int lane = tid & 31;

    for (int i = tid; i < STATS; i += THREADS) s_stats[i] = 0.0f;

    const long long wbase =
        (long long)blockIdx.x * BLOCK_ELEMS + (long long)wave * WAVE_ELEMS;
    const char* gp = (const char*)(y_pred + wbase);
    const char* gt = (const char*)(y_true + wbase);
    const char* gb = (const char*)(basin + wbase);

    auto issue = [&](int it, int buf) {
        const uint32_t base = lds_off(&s_stage[wave][buf][0]);
        const long long e = (long long)it * WAVE_TILE;
        async_b128(base +         lane * 16, gp + e * 4 + lane * 16);
        async_b128(base + 512  +  lane * 16, gt + e * 4 + lane * 16);
        async_b128(base + 1024 +  lane * 16, gb + e * 8 + lane * 16);
        async_b128(base + 1536 +  lane * 16, gb + e * 8 + 512 + lane * 16);
    };

    auto process = [&](int buf) {
        const char* sb = &s_stage[wave][buf][0];
        float4 p  = *(const float4*)(sb + lane * 16);
        float4 t  = *(const float4*)(sb + 512 + lane * 16);
        uint4  b0 = *(const uint4*)(sb + 1024 + lane * 32);
        uint4  b1 = *(const uint4*)(sb + 1024 + lane * 32 + 16);

        unsigned bs[4] = {b0.x, b0.z, b1.x, b1.z};  // int64 low words, ids < 671
        float    pv[4] = {p.x, p.y, p.z, p.w};
        float    tv[4] = {t.x, t.y, t.z, t.w};

#pragma unroll
        for (int e = 0; e < 4; ++e) {
            float y = tv[e];
            float d = y - pv[e];
            unsigned b = bs[e];
            atomicAdd(&s_stats[0 * NB + b], 1.0f);    // count
            atomicAdd(&s_stats[1 * NB + b], y);       // sum y
            atomicAdd(&s_stats[2 * NB + b], y * y);   // sum y^2
            atomicAdd(&s_stats[3 * NB + b], d * d);   // ss_res
        }
    };

    issue(0, 0);
    issue(1, 1);
    __syncthreads();  // stats zeroed before any atomics

#pragma unroll
    for (int it = 0; it < ITERS; ++it) {
        const int buf = it % PIPE;
        if (it + 2 < ITERS) {
            issue(it + 2, (it + 2) % PIPE);
            // 12 outstanding max; <=8 means batch `it` (in-order) has retired
            asm volatile("s_wait_asynccnt 8" ::: "memory");
        } else if (it + 1 < ITERS) {
            asm volatile("s_wait_asynccnt 4" ::: "memory");
        } else {
            asm volatile("s_wait_asynccnt 0" ::: "memory");
        }
        process(buf);
    }

    __syncthreads();
    for (int i = tid; i < STATS; i += THREADS)
        atomicAdd(&gstats[i], s_stats[i]);
}

__global__ void MeanNSE_zero_kernel(float* __restrict__ g) {
    int i = blockIdx.x * blockDim.x + threadIdx.x;
    if (i < STATS) g[i] = 0.0f;
}

// Scalar tail (N not multiple of BLOCK_ELEMS; unused for N = 2^24 but kept
// for generality).
__global__ void MeanNSE_tail_kernel(const float* __restrict__ y_pred,
                                    const float* __restrict__ y_true,
                                    const long long* __restrict__ basin,
                                    float* __restrict__ gstats,
                                    long long base, long long N) {
    long long i = base + (long long)blockIdx.x * blockDim.x + threadIdx.x;
    if (i < N) {
        float y = y_true[i];
        float d = y - y_pred[i];
        int b = (int)basin[i];
        atomicAdd(&gstats[0 * NB + b], 1.0f);
        atomicAdd(&gstats[1 * NB + b], y);
        atomicAdd(&gstats[2 * NB + b], y * y);
        atomicAdd(&gstats[3 * NB + b], d * d);
    }
}

__global__ __launch_bounds__(1024)
void MeanNSE_finalize_kernel(const float* __restrict__ g, float* __restrict__ out) {
    __shared__ float red[1024];
    int t = threadIdx.x;
    float nse = 0.0f;
    if (t < NB) {
        float c  = g[0 * NB + t];
        float sy = g[1 * NB + t];
        float s2 = g[2 * NB + t];
        float sr = g[3 * NB + t];
        float mean  = sy / c;
        float sstot = s2 - sy * mean;    // == sum((y - mean)^2) algebraically
        nse = 1.0f - sr / (sstot + EPS);
    }
    red[t] = nse;
    __syncthreads();
    for (int s = 512; s > 0; s >>= 1) {
        if (t < s) red[t] += red[t + s];
        __syncthreads();
    }
    if (t == 0) out[0] = red[0] / (float)NB;
}

extern "C" void kernel_launch(void* const* d_in, const int* in_sizes, int n_in,
                              void* d_out, int out_size, void* d_ws, size_t ws_size,
                              hipStream_t stream) {
    const float*     y_pred = (const float*)d_in[0];
    const float*     y_true = (const float*)d_in[1];
    const long long* basin  = (const long long*)d_in[2];
    float* gstats = (float*)d_ws;
    float* out    = (float*)d_out;
    const long long N = (long long)in_sizes[0];

    MeanNSE_zero_kernel<<<(STATS + 255) / 256, 256, 0, stream>>>(gstats);

    const long long nblocks = N / BLOCK_ELEMS;
    if (nblocks > 0) {
        MeanNSE_accum_kernel<<<(int)nblocks, THREADS, 0, stream>>>(
            y_pred, y_true, basin, gstats);
    }
    const long long base = nblocks * BLOCK_ELEMS;
    const long long rem  = N - base;
    if (rem > 0) {
        MeanNSE_tail_kernel<<<(int)((rem + 255) / 256), 256, 0, stream>>>(
            y_pred, y_true, basin, gstats, base, N);
    }

    MeanNSE_finalize_kernel<<<1, 1024, 0, stream>>>(gstats, out);
}